// HSA_GCN_27212912787479
// MI455X (gfx1250) — compile-verified
//
#include <hip/hip_runtime.h>
#include <hip/hip_bf16.h>

#define DFEAT 128

typedef float v2f __attribute__((ext_vector_type(2)));
typedef float v8f __attribute__((ext_vector_type(8)));

// ---------------------------------------------------------------- utilities
__global__ __launch_bounds__(256) void k_zero_i(int* __restrict__ p, int n) {
  int i = blockIdx.x * blockDim.x + threadIdx.x;
  if (i < n) p[i] = 0;
}

__global__ __launch_bounds__(256) void k_count_deg(const int* __restrict__ dst,
                                                   int* __restrict__ cnt, int e) {
  int i = blockIdx.x * blockDim.x + threadIdx.x;
  if (i < e) atomicAdd(&cnt[dst[i]], 1);
}

// dinv = (deg + 1)^-1/2   (+1 = self loop; always > 0)
__global__ __launch_bounds__(256) void k_dinv(const int* __restrict__ cnt,
                                              float* __restrict__ dinv, int n) {
  int i = blockIdx.x * blockDim.x + threadIdx.x;
  if (i < n) dinv[i] = rsqrtf((float)cnt[i] + 1.0f);
}

// ------------------------------------------------- exclusive scan (3 kernels)
// Valid for n <= 256*256 = 65536 (harness: N = 50000, nb = 196).
__global__ __launch_bounds__(256) void k_scan1(const int* __restrict__ in,
                                               int* __restrict__ out,
                                               int* __restrict__ bsum, int n) {
  __shared__ int sh[256];
  const int tid = threadIdx.x;
  const int i = blockIdx.x * 256 + tid;
  int v = (i < n) ? in[i] : 0;
  sh[tid] = v;
  __syncthreads();
#pragma unroll
  for (int off = 1; off < 256; off <<= 1) {
    int t = (tid >= off) ? sh[tid - off] : 0;
    __syncthreads();
    sh[tid] += t;
    __syncthreads();
  }
  if (i < n) out[i] = sh[tid] - v;  // exclusive
  if (tid == 255) bsum[blockIdx.x] = sh[255];
}

__global__ __launch_bounds__(256) void k_scan2(int* __restrict__ bsum, int nb) {
  __shared__ int sh[256];
  const int tid = threadIdx.x;
  int v = (tid < nb) ? bsum[tid] : 0;
  sh[tid] = v;
  __syncthreads();
#pragma unroll
  for (int off = 1; off < 256; off <<= 1) {
    int t = (tid >= off) ? sh[tid - off] : 0;
    __syncthreads();
    sh[tid] += t;
    __syncthreads();
  }
  if (tid < nb) bsum[tid] = sh[tid] - v;
}

__global__ __launch_bounds__(256) void k_scan3(int* __restrict__ rowstart,
                                               const int* __restrict__ bsum,
                                               int* __restrict__ cur, int n) {
  int i = blockIdx.x * 256 + threadIdx.x;
  if (i < n) {
    rowstart[i] += bsum[blockIdx.x];
    cur[i] = 0;
  }
}

// bucket-fill: csr_src[rowstart[dst] + k] = src  (k = atomic cursor)
__global__ __launch_bounds__(256) void k_fill_csr(const int* __restrict__ src,
                                                  const int* __restrict__ dst,
                                                  const int* __restrict__ rowstart,
                                                  int* __restrict__ cur,
                                                  int* __restrict__ csr_src, int e) {
  int i = blockIdx.x * blockDim.x + threadIdx.x;
  if (i < e) {
    int d = dst[i];
    int pos = rowstart[d] + atomicAdd(&cur[d], 1);
    csr_src[pos] = src[i];
  }
}

// ---------------------------------------------------------------- GEMM H = X @ W
// Block = 256 threads = 8 waves; block computes 16 rows x 128 cols of H.
// Wave w owns the 16x16 tile at columns [16w,16w+16); K = 128 in 32 steps of 4.
__global__ __launch_bounds__(256) void k_gemm_xw(const float* __restrict__ X,
                                                 const float* __restrict__ W,
                                                 float* __restrict__ H, int n) {
#if __has_builtin(__builtin_amdgcn_wmma_f32_16x16x4_f32)
  const int lane = threadIdx.x & 31;
  const int wave = threadIdx.x >> 5;
  const int rowBase = blockIdx.x * 16;
  const int m = lane & 15;            // M (for A) / N (for B,C,D) within tile
  const int khalf = (lane >> 4) * 2;  // lanes 16-31 hold upper K pair
  const int col = wave * 16 + m;

  int arow = rowBase + m;
  if (arow >= n) arow = n - 1;  // clamp (n is a multiple of 16 in this harness)
  const float* __restrict__ xrow = X + (size_t)arow * DFEAT;

  v8f c = {0.f, 0.f, 0.f, 0.f, 0.f, 0.f, 0.f, 0.f};
#pragma unroll
  for (int k0 = 0; k0 < DFEAT; k0 += 4) {
    v2f a, b;
    a.x = xrow[k0 + khalf + 0];
    a.y = xrow[k0 + khalf + 1];
    b.x = W[(size_t)(k0 + khalf + 0) * DFEAT + col];
    b.y = W[(size_t)(k0 + khalf + 1) * DFEAT + col];
    c = __builtin_amdgcn_wmma_f32_16x16x4_f32(false, a, false, b, (short)0, c,
                                              false, false);
  }
  const int rgrp = (lane >> 4) * 8;  // C/D: lanes 16-31 hold M = v + 8
#pragma unroll
  for (int v = 0; v < 8; ++v) {
    int r = rowBase + rgrp + v;
    if (r < n) H[(size_t)r * DFEAT + col] = c[v];
  }
#else
  // Scalar fallback (host pass / builtin unavailable). Same grid shape.
  const int rowBase = blockIdx.x * 16;
#pragma unroll 1
  for (int t = 0; t < 8; ++t) {
    int idx = threadIdx.x + t * 256;  // 0..2047
    int r = rowBase + (idx >> 7);
    int cc = idx & 127;
    if (r < n) {
      float acc = 0.f;
      for (int k = 0; k < DFEAT; ++k)
        acc = fmaf(X[(size_t)r * DFEAT + k], W[(size_t)k * DFEAT + cc], acc);
      H[(size_t)r * DFEAT + cc] = acc;
    }
  }
#endif
}

// ---------------------------------------------------------------- aggregation:
// wave per node; gather-based (no float atomics). Accumulator = float4/lane.
// AGG[d,:] = dinv[d] * sum_{s in N(d)} H[s,:]*dinv[s] + dinv[d]^2*H[d,:] + b[:]
__global__ __launch_bounds__(256) void k_aggregate(const float* __restrict__ H,
                                                   const float* __restrict__ dinv,
                                                   const float* __restrict__ b,
                                                   const int* __restrict__ rowstart,
                                                   const int* __restrict__ cnt,
                                                   const int* __restrict__ csr_src,
                                                   float* __restrict__ AGG, int n) {
  const int lane = threadIdx.x & 31;
  const int node = blockIdx.x * (blockDim.x >> 5) + (threadIdx.x >> 5);
  if (node >= n) return;
  const int start = rowstart[node];
  const int deg = cnt[node];

  float4 acc = {0.f, 0.f, 0.f, 0.f};
  int s_next = (deg > 0) ? csr_src[start] : 0;
  for (int j = 0; j < deg; ++j) {
    int s = s_next;
    if (j + 1 < deg) s_next = csr_src[start + j + 1];  // prefetch next index
    float w = dinv[s];
    float4 h = ((const float4*)(H + (size_t)s * DFEAT))[lane];
    acc.x = fmaf(h.x, w, acc.x);
    acc.y = fmaf(h.y, w, acc.y);
    acc.z = fmaf(h.z, w, acc.z);
    acc.w = fmaf(h.w, w, acc.w);
  }
  const float dd = dinv[node];
  const float sw = dd * dd;
  float4 hs = ((const float4*)(H + (size_t)node * DFEAT))[lane];
  float4 bv = ((const float4*)b)[lane];
  float4 o;
  o.x = fmaf(acc.x, dd, fmaf(hs.x, sw, bv.x));
  o.y = fmaf(acc.y, dd, fmaf(hs.y, sw, bv.y));
  o.z = fmaf(acc.z, dd, fmaf(hs.z, sw, bv.z));
  o.w = fmaf(acc.w, dd, fmaf(hs.w, sw, bv.w));
  ((float4*)(AGG + (size_t)node * DFEAT))[lane] = o;
}

// ---------------------------------------------------------------- BN stats
// Per-block partials (deterministic tree reduce; no float atomics).
// part[b*256 + c] = partial sum of column c; part[b*256 + 128 + c] = sumsq.
__global__ __launch_bounds__(256) void k_col_stats_part(const float* __restrict__ A,
                                                        float* __restrict__ part,
                                                        int n) {
  __shared__ float sh[512];
  const int c = threadIdx.x & 127;
  const int rg = threadIdx.x >> 7;  // 0 or 1
  float s = 0.f, s2 = 0.f;
  for (int row = blockIdx.x * 2 + rg; row < n; row += gridDim.x * 2) {
    float v = A[(size_t)row * DFEAT + c];
    s += v;
    s2 = fmaf(v, v, s2);
  }
  sh[threadIdx.x] = s;
  sh[256 + threadIdx.x] = s2;
  __syncthreads();
  if (threadIdx.x < 128) {
    part[blockIdx.x * 256 + c] = sh[c] + sh[c + 128];
    part[blockIdx.x * 256 + 128 + c] = sh[256 + c] + sh[384 + c];
  }
}

__global__ __launch_bounds__(256) void k_stats_reduce(const float* __restrict__ part,
                                                      float* __restrict__ stats,
                                                      int nblk) {
  int t = threadIdx.x;  // 0..255 -> stats[0:128]=sum, stats[128:256]=sumsq
  float acc = 0.f;
  for (int b2 = 0; b2 < nblk; ++b2) acc += part[b2 * 256 + t];
  stats[t] = acc;
}

// scale = gamma * rsqrt(var + eps); shift = beta - mean*scale
__global__ __launch_bounds__(128) void k_bn_finalize(const float* __restrict__ stats,
                                                     const float* __restrict__ gamma,
                                                     const float* __restrict__ beta,
                                                     float* __restrict__ ss, float n) {
  int c = threadIdx.x;
  float mean = stats[c] / n;
  float var = stats[128 + c] / n - mean * mean;
  float sc = gamma[c] * rsqrtf(var + 1e-5f);
  ss[c] = sc;
  ss[128 + c] = fmaf(-mean, sc, beta[c]);
}

// Y = relu(A * scale + shift)
__global__ __launch_bounds__(256) void k_bn_apply_relu(const float* __restrict__ A,
                                                       const float* __restrict__ ss,
                                                       float* __restrict__ Y, int n) {
  int idx = blockIdx.x * blockDim.x + threadIdx.x;  // over n*32 float4s
  if (idx >= n * 32) return;
  int q = idx & 31;
  float4 sc = ((const float4*)ss)[q];
  float4 sh = ((const float4*)ss)[32 + q];
  float4 a = ((const float4*)A)[idx];
  float4 o;
  o.x = fmaxf(fmaf(a.x, sc.x, sh.x), 0.f);
  o.y = fmaxf(fmaf(a.y, sc.y, sh.y), 0.f);
  o.z = fmaxf(fmaf(a.z, sc.z, sh.z), 0.f);
  o.w = fmaxf(fmaf(a.w, sc.w, sh.w), 0.f);
  ((float4*)Y)[idx] = o;
}

// ---------------------------------------------------------------- launcher
extern "C" void kernel_launch(void* const* d_in, const int* in_sizes, int n_in,
                              void* d_out, int out_size, void* d_ws, size_t ws_size,
                              hipStream_t stream) {
  const float* x = (const float*)d_in[0];
  const int* ei = (const int*)d_in[1];
  const float* W1 = (const float*)d_in[2];
  const float* b1 = (const float*)d_in[3];
  const float* g1 = (const float*)d_in[4];
  const float* be1 = (const float*)d_in[5];
  const float* W2 = (const float*)d_in[6];
  const float* b2 = (const float*)d_in[7];
  const float* g2 = (const float*)d_in[8];
  const float* be2 = (const float*)d_in[9];

  const int N = in_sizes[0] / DFEAT;
  const int E = in_sizes[1] / 2;
  const int* src = ei;
  const int* dst = ei + E;

  // ---- workspace carve (256B-aligned regions)
  char* wsb = (char*)d_ws;
  auto carve = [&](size_t bytes) {
    char* p = wsb;
    wsb += (bytes + 255) & ~(size_t)255;
    return p;
  };
  const size_t nd = (size_t)N * DFEAT;
  int* cnt = (int*)carve((size_t)N * 4);
  int* cur = (int*)carve((size_t)N * 4);
  int* rowstart = (int*)carve((size_t)N * 4);
  int* bsum = (int*)carve(256 * 4);
  int* csr_src = (int*)carve((size_t)E * 4);
  float* dinv = (float*)carve((size_t)N * 4);
  float* H = (float*)carve(nd * 4);
  float* AGG = (float*)carve(nd * 4);
  float* Y1 = (float*)carve(nd * 4);
  float* part = (float*)carve(128 * 256 * 4);
  float* stats = (float*)carve(256 * 4);
  float* ss = (float*)carve(256 * 4);
  float* out = (float*)d_out;

  const int q4 = N * 32;  // float4 count of an N x 128 matrix
  const int nb = (N + 255) / 256;
  dim3 blk(256);
  dim3 gN(nb), gE((E + 255) / 256), gQ((q4 + 255) / 256);
  dim3 gGemm((N + 15) / 16), gAgg((N + 7) / 8), gStat(128);

  // ---- degree, dinv, CSR build (counting sort by dst)
  k_zero_i<<<gN, blk, 0, stream>>>(cnt, N);
  k_count_deg<<<gE, blk, 0, stream>>>(dst, cnt, E);
  k_dinv<<<gN, blk, 0, stream>>>(cnt, dinv, N);
  k_scan1<<<gN, blk, 0, stream>>>(cnt, rowstart, bsum, N);
  k_scan2<<<dim3(1), blk, 0, stream>>>(bsum, nb);
  k_scan3<<<gN, blk, 0, stream>>>(rowstart, bsum, cur, N);
  k_fill_csr<<<gE, blk, 0, stream>>>(src, dst, rowstart, cur, csr_src, E);

  // ---- layer 1
  k_gemm_xw<<<gGemm, blk, 0, stream>>>(x, W1, H, N);
  k_aggregate<<<gAgg, blk, 0, stream>>>(H, dinv, b1, rowstart, cnt, csr_src, AGG, N);
  k_col_stats_part<<<gStat, blk, 0, stream>>>(AGG, part, N);
  k_stats_reduce<<<dim3(1), blk, 0, stream>>>(part, stats, 128);
  k_bn_finalize<<<dim3(1), dim3(128), 0, stream>>>(stats, g1, be1, ss, (float)N);
  k_bn_apply_relu<<<gQ, blk, 0, stream>>>(AGG, ss, Y1, N);

  // ---- layer 2
  k_gemm_xw<<<gGemm, blk, 0, stream>>>(Y1, W2, H, N);
  k_aggregate<<<gAgg, blk, 0, stream>>>(H, dinv, b2, rowstart, cnt, csr_src, AGG, N);
  k_col_stats_part<<<gStat, blk, 0, stream>>>(AGG, part, N);
  k_stats_reduce<<<dim3(1), blk, 0, stream>>>(part, stats, 128);
  k_bn_finalize<<<dim3(1), dim3(128), 0, stream>>>(stats, g2, be2, ss, (float)N);
  k_bn_apply_relu<<<gQ, blk, 0, stream>>>(AGG, ss, out, N);
}